// Encoder_28595892256973
// MI455X (gfx1250) — compile-verified
//
#include <hip/hip_runtime.h>

// ---------------------------------------------------------------------------
// Tree encoder for MI455X (gfx1250, wave32, WMMA).
//
// 14 serially-dependent stages, ~16 GFLOP total, weights L2-resident ->
// latency-bound. One fused kernel per layer:
//   stage1: build swapped-children row block H[16 x 2*idim] in LDS
//   stage2: direction-gated GEMM via V_WMMA_F32_16X16X4_F32 (+bias),
//           register-resident gate select (v_cndmask), PReLU
//   stage3: dense skip GEMM (concat [U | subsampled backup]) via WMMA
// fp32 WMMA keeps the reference dtype exactly. Inner K-loops are software-
// pipelined (depth 1) so the s_wait before each WMMA covers loads issued a
// full iteration earlier instead of stalling on just-issued loads.
// ---------------------------------------------------------------------------

typedef __attribute__((ext_vector_type(2))) float v2f;
typedef __attribute__((ext_vector_type(8))) float v8f;

#define NDIR   7
#define WAVES  8
#define TPB    (WAVES * 32)
#define PAD    4            // LDS row padding (floats) to break bank conflicts
#define CTRL   36           // floats reserved at smem base for gate/rev/mask

// ---------------------------------------------------------------------------
// Leaf: x0[n,o] = prelu(points[n,:3] @ W[3,8] + b[8], a) ; one thread/element.
// ---------------------------------------------------------------------------
__global__ __launch_bounds__(256) void leaf_kernel(
    const float* __restrict__ pts,   // (B*N, 3)
    const float* __restrict__ W,     // (3, 8)
    const float* __restrict__ bias,  // (8)
    const float* __restrict__ aP,    // scalar
    float* __restrict__ xout,        // (B*N, 8)
    int total)
{
  int t = blockIdx.x * blockDim.x + threadIdx.x;
  int n = t >> 3, o = t & 7;
  if (n >= total) return;
  float acc = bias[o]
            + pts[n * 3 + 0] * W[0 * 8 + o]
            + pts[n * 3 + 1] * W[1 * 8 + o]
            + pts[n * 3 + 2] * W[2 * 8 + o];
  float alpha = *aP;
  xout[n * 8 + o] = (acc >= 0.f) ? acc : alpha * acc;
}

// ---------------------------------------------------------------------------
// One merge layer. Workgroup = 16 nodes (one WMMA M-tile), 8 wave32 waves.
// ---------------------------------------------------------------------------
__global__ __launch_bounds__(TPB) void tree_layer_kernel(
    const float* __restrict__ xprev,  // (B*(2*npb), idim)
    const float* __restrict__ xskip,  // (B*(8*npb), sdim) or nullptr
    float* __restrict__ xout,         // (B*npb, odim)
    const int* __restrict__ vec,      // (B*npb)
    const int* __restrict__ dmap,     // (NDIR)
    const int* __restrict__ drev,     // (NDIR)
    const float* __restrict__ Wg,     // (NDIR, 2*idim, odim)
    const float* __restrict__ bg,     // (NDIR, odim)
    const float* __restrict__ Ws,     // (odim+sdim, odim) or nullptr
    const float* __restrict__ bs,     // (odim) or nullptr
    const float* __restrict__ aP,     // scalar PReLU slope
    int idim, int odim, int sdim, int npb)
{
  const int K1  = 2 * idim;
  const int K1p = K1 + PAD;
  const int K2  = odim + sdim;        // concat K for skip GEMM
  const int K2p = K2 + PAD;

  extern __shared__ float smem[];
  int*   gate_s = (int*)smem;         // 16  (16B aligned)
  int*   rev_s  = gate_s + 16;        // 16
  int*   mask_s = rev_s + 16;         // 1 (+3 pad)
  float* H      = smem + CTRL;        // 16 x K1p (16B aligned rows)
  float* A2     = H + 16 * K1p;       // 16 x K2p : [U | SG] for stage 3

  const int tid  = threadIdx.x;
  const int base = blockIdx.x * 16;   // first node of this tile

  if (tid < 16) {
    int v = vec[base + tid];
    gate_s[tid] = dmap[v];
    rev_s[tid]  = drev[v];
  }
  __syncthreads();

  // ---- stage 1: H = swapped children; SG rows of A2 ----
  for (int idx = tid; idx < 16 * K1; idx += TPB) {
    int r = idx / K1;
    int i = idx - r * K1;
    int n = base + r;
    int b = n / npb;
    int j = n - b * npb;
    int half = (i >= idim) ? 1 : 0;
    int off  = half ? (i - idim) : i;
    int ch   = rev_s[r] ^ half;       // rev=0: [lch|rch], rev=1: [rch|lch]
    int pn   = b * (npb << 1) + (j << 1) + ch;
    H[r * K1p + i] = xprev[(size_t)pn * idim + off];
  }
  if (xskip) {
    for (int idx = tid; idx < 16 * sdim; idx += TPB) {
      int r  = idx / sdim;
      int s2 = idx - r * sdim;
      int n  = base + r;
      int b  = n / npb;
      int j  = n - b * npb;
      int sn = b * (npb << 3) + (j << 3);   // stride-8 subsample (SAMPLE=3)
      A2[r * K2p + odim + s2] = xskip[(size_t)sn * sdim + s2];
    }
  }
  if (tid == 0) {                     // gates present in this tile
    int m = 0;
    for (int r = 0; r < 16; ++r) m |= 1 << gate_s[r];
    *mask_s = m;
  }
  __syncthreads();

  const int wave   = tid >> 5;
  const int lane   = tid & 31;
  const int half   = lane >> 4;       // frag layout: lanes 16-31 hold K+2
  const int lane16 = lane & 15;
  const float alpha = *aP;
  const int gm  = *mask_s;
  const int nct = odim >> 4;          // 16-wide column tiles
  const int rowOff1 = lane16 * K1p;

  // this lane's 8 C-rows (M = v + 8*half) and their gates, in registers
  int gt[8];
  {
    int4 g0 = *(const int4*)&gate_s[half << 3];
    int4 g1 = *(const int4*)&gate_s[(half << 3) + 4];
    gt[0]=g0.x; gt[1]=g0.y; gt[2]=g0.z; gt[3]=g0.w;
    gt[4]=g1.x; gt[5]=g1.y; gt[6]=g1.z; gt[7]=g1.w;
  }

  // ---- stage 2: gated GEMM, register gate-select, PReLU ----
  for (int ct = wave; ct < nct; ct += WAVES) {
    const int col = (ct << 4) + lane16;
    v8f out;
    #pragma unroll
    for (int v = 0; v < 8; ++v) out[v] = 0.f;

    for (int g = 0; g < NDIR; ++g) {
      if (!(gm & (1 << g))) continue;          // wave-uniform skip
      const float bgv = bg[g * odim + col];
      v8f c;
      #pragma unroll
      for (int v = 0; v < 8; ++v) c[v] = bgv;  // C starts at bias
      const float* bcol = Wg + (size_t)g * K1 * odim + col;

      // depth-1 software pipeline over K (K1 >= 16, multiple of 4)
      int kb = half << 1;                       // kk=0 fragment
      v2f a  = *(const v2f*)&H[rowOff1 + kb];
      float bx = bcol[(size_t)kb * odim];
      float by = bcol[(size_t)(kb + 1) * odim];
      for (int kk = 4; kk < K1; kk += 4) {
        const int kn = kk + (half << 1);
        v2f an  = *(const v2f*)&H[rowOff1 + kn];
        float bxn = bcol[(size_t)kn * odim];
        float byn = bcol[(size_t)(kn + 1) * odim];
        v2f bf; bf.x = bx; bf.y = by;
        c = __builtin_amdgcn_wmma_f32_16x16x4_f32(
                false, a, false, bf, (short)0, c, false, false);
        a = an; bx = bxn; by = byn;
      }
      v2f bf; bf.x = bx; bf.y = by;
      c = __builtin_amdgcn_wmma_f32_16x16x4_f32(
              false, a, false, bf, (short)0, c, false, false);

      #pragma unroll
      for (int v = 0; v < 8; ++v)               // v_cmp + v_cndmask, no branch
        out[v] = (gt[v] == g) ? c[v] : out[v];
    }

    #pragma unroll
    for (int v = 0; v < 8; ++v) {
      const int rowm = v + (half << 3);
      float y = out[v];
      y = (y >= 0.f) ? y : alpha * y;
      if (xskip) A2[rowm * K2p + col] = y;      // feed concat for stage 3
      else       xout[(size_t)(base + rowm) * odim + col] = y;
    }
  }

  if (!xskip) return;                 // uniform across block
  __syncthreads();

  // ---- stage 3: dense skip GEMM over concat [U | SG], PReLU, writeout ----
  const int rowOff2 = lane16 * K2p;
  for (int ct = wave; ct < nct; ct += WAVES) {
    const int col = (ct << 4) + lane16;
    const float bsv = bs[col];
    v8f c;
    #pragma unroll
    for (int v = 0; v < 8; ++v) c[v] = bsv;
    const float* bcol = Ws + col;

    int kb = half << 1;
    v2f a  = *(const v2f*)&A2[rowOff2 + kb];
    float bx = bcol[(size_t)kb * odim];
    float by = bcol[(size_t)(kb + 1) * odim];
    for (int kk = 4; kk < K2; kk += 4) {
      const int kn = kk + (half << 1);
      v2f an  = *(const v2f*)&A2[rowOff2 + kn];
      float bxn = bcol[(size_t)kn * odim];
      float byn = bcol[(size_t)(kn + 1) * odim];
      v2f bf; bf.x = bx; bf.y = by;
      c = __builtin_amdgcn_wmma_f32_16x16x4_f32(
              false, a, false, bf, (short)0, c, false, false);
      a = an; bx = bxn; by = byn;
    }
    v2f bf; bf.x = bx; bf.y = by;
    c = __builtin_amdgcn_wmma_f32_16x16x4_f32(
            false, a, false, bf, (short)0, c, false, false);

    #pragma unroll
    for (int v = 0; v < 8; ++v) {
      const int rowm = v + (half << 3);
      float y = c[v];
      y = (y >= 0.f) ? y : alpha * y;
      xout[(size_t)(base + rowm) * odim + col] = y;
    }
  }
}

// ---------------------------------------------------------------------------
// Host: 1 leaf launch + 13 layer launches, stream-ordered (graph-capture safe).
// ---------------------------------------------------------------------------
extern "C" void kernel_launch(void* const* d_in, const int* in_sizes, int n_in,
                              void* d_out, int out_size, void* d_ws, size_t ws_size,
                              hipStream_t stream)
{
  (void)in_sizes; (void)n_in; (void)out_size; (void)ws_size;
  const int B = 32, N = 8192;
  static const int OD[14] = {8,16,32,64,128,256,512,512,512,512,512,512,512,512};

  const float* points = (const float*)d_in[0];
  const int*   dmap   = (const int*)d_in[1];
  const int*   drev   = (const int*)d_in[2];
  const int*   vec[14];
  for (int k = 1; k <= 13; ++k) vec[k] = (const int*)d_in[2 + k];

  // params flattened in JAX tree order: sorted dict keys => 'layers' < 'leaf';
  // per layer dict sorted: Wg, [Ws], a, bg, [bs].
  const float *Wg[14], *WsP[14], *aL[14], *bgP[14], *bsP[14];
  int pi = 16;
  for (int k = 1; k <= 13; ++k) {
    Wg[k]  = (const float*)d_in[pi++];
    WsP[k] = nullptr; bsP[k] = nullptr;
    if (k >= 3) WsP[k] = (const float*)d_in[pi++];
    aL[k]  = (const float*)d_in[pi++];
    bgP[k] = (const float*)d_in[pi++];
    if (k >= 3) bsP[k] = (const float*)d_in[pi++];
  }
  const float* leafW = (const float*)d_in[pi++];   // leaf: W, a, b (sorted)
  const float* leafA = (const float*)d_in[pi++];
  const float* leafB = (const float*)d_in[pi++];

  // 4-slot activation ring (skip reaches back 3 layers); 8 MB per slot.
  const size_t SLOT = (size_t)2 * 1024 * 1024;     // floats
  float* ws = (float*)d_ws;
  float* slot[4] = { ws, ws + SLOT, ws + 2 * SLOT, ws + 3 * SLOT };

  {
    int total = B * N;
    leaf_kernel<<<dim3((total * 8) / 256), 256, 0, stream>>>(
        points, leafW, leafB, leafA, slot[0], total);
  }

  for (int k = 1; k <= 13; ++k) {
    const int idim = OD[k - 1], odim = OD[k];
    const int sdim = (k >= 3) ? OD[k - 3] : 0;
    const int npb  = N >> k;
    const int nodes = B * npb;                     // always a multiple of 16
    const float* xprev = slot[(k - 1) & 3];
    const float* xskip = (k >= 3) ? slot[(k - 3) & 3] : nullptr;
    float* xout = (k == 13) ? (float*)d_out : slot[k & 3];

    const int K1p = 2 * idim + PAD;
    const int K2p = odim + sdim + PAD;
    const size_t smem = (size_t)(CTRL + 16 * (K1p + K2p)) * sizeof(float);

    tree_layer_kernel<<<dim3(nodes / 16), TPB, smem, stream>>>(
        xprev, xskip, xout, vec[k], dmap, drev,
        Wg[k], bgP[k], WsP[k], bsP[k], aL[k],
        idim, odim, sdim, npb);
  }
}